// WeightedMSELoss_6184752906812
// MI455X (gfx1250) — compile-verified
//
#include <hip/hip_runtime.h>

// WeightedMSELoss on MI455X (gfx1250, wave32).
// HBM-bound streaming reduction: 134 MB in, 4 B out -> ~5.8 us floor @ 23.3 TB/s.
// Strategy: float4 (global_load_b128) streaming, LDS reciprocal-weight LUT,
// deterministic hierarchical sum. Wave-level reduction uses V_WMMA_F32_16X16X4_F32
// (A = ones) to fold 32 lane-partials into column sums in one matrix op.

typedef __attribute__((ext_vector_type(2))) float v2f;
typedef __attribute__((ext_vector_type(8))) float v8f;

#define NBLOCKS  1024
#define NTHREADS 256
#define NWAVES   (NTHREADS / 32)
#define MAX_W    512

// Reduce one wave's 32 partial sums. A = ones(16x4) f32, B = {s, 0} per lane.
// D[i][j] = sum_k B[k][j]  (column sums of B). c[0] holds the 16 column sums in
// lanes 0-15 (row M=0) and the same sums again in lanes 16-31 (row M=8), so a
// full 32-lane butterfly over c[0] yields 2x the wave total (layout-independent:
// every B element contributes to exactly one column sum).
__device__ __forceinline__ float wave_reduce_wmma(float s) {
    v2f a; a.x = 1.0f; a.y = 1.0f;   // all-ones 16x4 A matrix
    v2f b; b.x = s;    b.y = 0.0f;   // 64-slot B: one partial per lane
    v8f c = {};
    c = __builtin_amdgcn_wmma_f32_16x16x4_f32(
            /*neg_a=*/false, a, /*neg_b=*/false, b,
            /*c_mod=*/(short)0, c, /*reuse_a=*/false, /*reuse_b=*/false);
    float x = c[0];
#pragma unroll
    for (int off = 16; off >= 1; off >>= 1)
        x += __shfl_xor(x, off, 32);
    return x * 0.5f;                 // butterfly double-counts the duplicated rows
}

__device__ __forceinline__ float term(float p, float t, const float* lw, int nw_m1) {
    int idx = (int)(t * 500.0f);             // v_cvt_i32_f32: trunc == astype(int32)
    idx = min(max(idx, 0), nw_m1);
    float w = lw[idx];                       // ds_load_b32 gather from LDS LUT
    float d = p - t;                         // |.|^2 == (.)^2
    return w * d * d;
}

__global__ void __launch_bounds__(NTHREADS)
wmse_partial(const float* __restrict__ pred, const float* __restrict__ truth,
             const float* __restrict__ wtab, float* __restrict__ partials,
             int n, int nw) {
    __shared__ float lw[MAX_W];
    __shared__ float lsum[NWAVES];
    const int tid = threadIdx.x;

    // Build reciprocal weight LUT once per block (<=512 entries, negligible).
    const int nwc = min(nw, MAX_W);
    for (int i = tid; i < nwc; i += NTHREADS) {
        float w = wtab[i];
        lw[i] = (w < 1e-5f) ? 1.0f : (1.0f / w);
    }
    __syncthreads();

    const int nw_m1  = nwc - 1;
    const int n4     = n >> 2;
    const int stride = NBLOCKS * NTHREADS;
    const int gtid   = blockIdx.x * NTHREADS + tid;
    const float4* __restrict__ p4 = (const float4*)pred;
    const float4* __restrict__ t4 = (const float4*)truth;

    float s0 = 0.0f, s1 = 0.0f;              // two accumulators to hide latency
    for (int i = gtid; i < n4; i += stride) {
        float4 p = p4[i];                    // global_load_b128
        float4 t = t4[i];                    // global_load_b128
        s0 += term(p.x, t.x, lw, nw_m1);
        s1 += term(p.y, t.y, lw, nw_m1);
        s0 += term(p.z, t.z, lw, nw_m1);
        s1 += term(p.w, t.w, lw, nw_m1);
    }
    // Scalar tail (n not multiple of 4) — no-op for the reference's 2^23.
    const int base = n4 << 2;
    if (gtid < (n - base))
        s0 += term(pred[base + gtid], truth[base + gtid], lw, nw_m1);

    // Wave reduce via f32 WMMA (EXEC all ones here: straight-line, reconverged).
    float ws = wave_reduce_wmma(s0 + s1);
    if ((tid & 31) == 0) lsum[tid >> 5] = ws;
    __syncthreads();
    if (tid == 0) {
        float b = 0.0f;
#pragma unroll
        for (int w = 0; w < NWAVES; ++w) b += lsum[w];   // fixed order: deterministic
        partials[blockIdx.x] = b;
    }
}

__global__ void __launch_bounds__(NTHREADS)
wmse_final(const float* __restrict__ partials, float* __restrict__ out,
           int nblocks, float scale) {
    __shared__ float lsum[NWAVES];
    const int tid = threadIdx.x;
    float s = 0.0f;
    for (int i = tid; i < nblocks; i += NTHREADS) s += partials[i];
    float ws = wave_reduce_wmma(s);
    if ((tid & 31) == 0) lsum[tid >> 5] = ws;
    __syncthreads();
    if (tid == 0) {
        float b = 0.0f;
#pragma unroll
        for (int w = 0; w < NWAVES; ++w) b += lsum[w];
        out[0] = b * scale;                  // scale = 100/N (exact: 25 * 2^-21)
    }
}

extern "C" void kernel_launch(void* const* d_in, const int* in_sizes, int n_in,
                              void* d_out, int out_size, void* d_ws, size_t ws_size,
                              hipStream_t stream) {
    const float* pred  = (const float*)d_in[0];
    const float* truth = (const float*)d_in[1];
    const float* wtab  = (const float*)d_in[2];
    const int n  = in_sizes[0];
    const int nw = in_sizes[2];

    float* partials = (float*)d_ws;          // 1024 floats = 4 KB scratch

    wmse_partial<<<NBLOCKS, NTHREADS, 0, stream>>>(pred, truth, wtab, partials, n, nw);

    const float scale = 100.0f / (float)n;
    wmse_final<<<1, NTHREADS, 0, stream>>>(partials, (float*)d_out, NBLOCKS, scale);
}